// GeometricKernelAttnOp_9371618640644
// MI455X (gfx1250) — compile-verified
//
#include <hip/hip_runtime.h>

// Static problem geometry (from the reference): levels 128x128, 64x64, 32x32, 16x16
#define NB 2
#define QQ 8192
#define HH 8
#define CC 32
#define SS 21760   // 16384 + 4096 + 1024 + 256

typedef __attribute__((ext_vector_type(2))) float v2f;
typedef __attribute__((ext_vector_type(8))) float v8f;

__device__ __forceinline__ float rdlane_f(float v, int l) {
    return __int_as_float(__builtin_amdgcn_readlane(__float_as_int(v), l));
}

__global__ __launch_bounds__(256)
void msda_fwd_kernel(const float* __restrict__ value,
                     const float* __restrict__ loc,
                     const float* __restrict__ attw,
                     float* __restrict__ out)
{
    const int lane = threadIdx.x & 31;
    const int wave = blockIdx.x * (blockDim.x >> 5) + (threadIdx.x >> 5);
    // wave -> (n, q, h); all uniform within the wave
    const int h = wave & (HH - 1);
    const int q = (wave / HH) & (QQ - 1);
    const int n = wave / (HH * QQ);
    const int qbase = (n * QQ + q) * HH + h;

    // ---- per-lane sample-point setup: lane (0..15) owns point pt = l*4+p ----
    const int pt  = lane & 15;
    const int lvl = pt >> 2;
    const int Wl  = 128 >> lvl;                              // Hl == Wl
    const int start = (65536 - (65536 >> (2 * lvl))) / 3;    // 0,16384,20480,21504

    const float lx = __builtin_nontemporal_load(&loc[qbase * 32 + 2 * pt]);
    const float ly = __builtin_nontemporal_load(&loc[qbase * 32 + 2 * pt + 1]);
    const float aw = __builtin_nontemporal_load(&attw[qbase * 16 + pt]);

    const float fw = (float)Wl;
    const float x = fmaf(lx, fw, -0.5f);
    const float y = fmaf(ly, fw, -0.5f);
    const float fx0 = floorf(x), fy0 = floorf(y);
    const float dx = x - fx0, dy = y - fy0;
    const int x0 = (int)fx0, y0 = (int)fy0;
    const int x1 = x0 + 1,  y1 = y0 + 1;

    const bool vx0 = ((unsigned)x0 < (unsigned)Wl);
    const bool vx1 = ((unsigned)x1 < (unsigned)Wl);
    const bool vy0 = ((unsigned)y0 < (unsigned)Wl);
    const bool vy1 = ((unsigned)y1 < (unsigned)Wl);

    const float wy0 = aw * (1.0f - dy), wy1 = aw * dy;
    const float wx0 = 1.0f - dx,        wx1 = dx;
    const float w00 = (vy0 && vx0) ? wy0 * wx0 : 0.0f;
    const float w01 = (vy0 && vx1) ? wy0 * wx1 : 0.0f;
    const float w10 = (vy1 && vx0) ? wy1 * wx0 : 0.0f;
    const float w11 = (vy1 && vx1) ? wy1 * wx1 : 0.0f;

    const int cx0 = min(max(x0, 0), Wl - 1);
    const int cx1 = min(max(x1, 0), Wl - 1);
    const int cy0 = min(max(y0, 0), Wl - 1);
    const int cy1 = min(max(y1, 0), Wl - 1);

    // element offsets into value (N,S,H,C): ((n*S + s)*H + h)*C
    const int sb = n * SS + start;
    const int r0 = sb + cy0 * Wl;
    const int r1 = sb + cy1 * Wl;
    const int hb = h * CC;
    const int o00 = (r0 + cx0) * (HH * CC) + hb;
    const int o01 = (r0 + cx1) * (HH * CC) + hb;
    const int o10 = (r1 + cx0) * (HH * CC) + hb;
    const int o11 = (r1 + cx1) * (HH * CC) + hb;

    // ---- gather + accumulate; one partial accumulator per level ----
    float acc[4] = {0.0f, 0.0f, 0.0f, 0.0f};
#pragma unroll
    for (int p = 0; p < 16; ++p) {
        const int   s00 = __builtin_amdgcn_readlane(o00, p);
        const int   s01 = __builtin_amdgcn_readlane(o01, p);
        const int   s10 = __builtin_amdgcn_readlane(o10, p);
        const int   s11 = __builtin_amdgcn_readlane(o11, p);
        const float g00 = rdlane_f(w00, p);
        const float g01 = rdlane_f(w01, p);
        const float g10 = rdlane_f(w10, p);
        const float g11 = rdlane_f(w11, p);
        // coalesced 128B wave loads: lane = channel
        const float v00 = value[s00 + lane];
        const float v01 = value[s01 + lane];
        const float v10 = value[s10 + lane];
        const float v11 = value[s11 + lane];
        acc[p >> 2] = fmaf(g00, v00,
                      fmaf(g01, v01,
                      fmaf(g10, v10,
                      fmaf(g11, v11, acc[p >> 2]))));
    }

    // ---- exact level-partial reduction on the matrix pipe ----
    // D = ones(16x4) * B + 0 with V_WMMA_F32_16X16X4_F32: every row of D equals
    // the column-wise sum of B's 4 rows, so the result is invariant to which
    // K-slot each partial occupies. B columns follow the documented lane
    // striping (n = lane for lanes 0-15, n = lane-16 for lanes 16-31).
    const float s0 = __shfl_xor(acc[0], 16, 32);
    const float s1 = __shfl_xor(acc[1], 16, 32);
    const float s2 = __shfl_xor(acc[2], 16, 32);
    const float s3 = __shfl_xor(acc[3], 16, 32);
    const bool lo = lane < 16;

    v2f ones; ones[0] = 1.0f; ones[1] = 1.0f;
    v8f cz = {0.0f, 0.0f, 0.0f, 0.0f, 0.0f, 0.0f, 0.0f, 0.0f};

    v2f blo; blo[0] = lo ? acc[0] : s1;  blo[1] = lo ? acc[2] : s3;   // channels 0-15
    v2f bhi; bhi[0] = lo ? s0 : acc[1];  bhi[1] = lo ? s2 : acc[3];   // channels 16-31

    v8f dlo = __builtin_amdgcn_wmma_f32_16x16x4_f32(
        false, ones, false, blo, (short)0, cz, false, false);
    v8f dhi = __builtin_amdgcn_wmma_f32_16x16x4_f32(
        false, ones, false, bhi, (short)0, cz, false, false);

    // D VGPR0: lanes 0-15 = row0 cols 0-15; lanes 16-31 = row8 cols 0-15
    // (all rows equal) -> lane l >= 16 of dhi[0] holds channel l exactly.
    const float res = lo ? dlo[0] : dhi[0];

    __builtin_nontemporal_store(res, &out[qbase * CC + lane]);
}

extern "C" void kernel_launch(void* const* d_in, const int* in_sizes, int n_in,
                              void* d_out, int out_size, void* d_ws, size_t ws_size,
                              hipStream_t stream) {
    (void)in_sizes; (void)n_in; (void)out_size; (void)d_ws; (void)ws_size;
    const float* value = (const float*)d_in[0];  // (N,S,H,C) f32
    // d_in[1], d_in[2] are static spatial shapes / level starts: hardcoded
    const float* loc   = (const float*)d_in[3];  // (N,Q,H,L,P,2) f32
    const float* attw  = (const float*)d_in[4];  // (N,Q,H,L,P) f32
    float* out = (float*)d_out;                  // (N,Q,H,C) f32

    const int total_waves = NB * QQ * HH;        // 131072
    const int waves_per_block = 256 / 32;        // 8
    dim3 grid(total_waves / waves_per_block);    // 16384 full blocks
    msda_fwd_kernel<<<grid, 256, 0, stream>>>(value, loc, attw, out);
}